// FeatureNormalization_70119636074512
// MI455X (gfx1250) — compile-verified
//
#include <hip/hip_runtime.h>
#include <stdint.h>

// Feature normalization, B=64 L=4096 C=128 fp32.
// Memory-bound (~335 MB HBM traffic ~= 15us @ 23.3 TB/s). Pass 1 streams
// frames through LDS with CDNA5 async global->LDS copies (ASYNCcnt,
// s_wait_asynccnt) double-buffered; pass 3 uses nontemporal b128 stores so the
// write-once output does not evict feat from the 192MB L2.

typedef float v4f __attribute__((ext_vector_type(4)));

#define Bn 64
#define Ln 4096
#define Cn 128
#define SPLIT 16
#define FPB (Ln / SPLIT) /* 256 frames per reduction block */
#define FPI 8            /* frames per iteration (one per wave) */

// gfx1250 async copy: copies 16 bytes at saddr+vaddr into LDS at vdst, counted by ASYNCcnt
__device__ __forceinline__ void async_ld_b128(unsigned lds_off, unsigned voff,
                                              unsigned long long sbase) {
  asm volatile("global_load_async_to_lds_b128 %0, %1, %2"
               :: "v"(lds_off), "v"(voff), "s"(sbase)
               : "memory");
}
#define WAIT_ASYNC(n) asm volatile("s_wait_asynccnt " #n ::: "memory")

// ---------------------------------------------------------------------------
// Pass 1: per-(b, slice) partial sum / sum-of-squares over valid frames.
// Grid: Bn*SPLIT blocks, 256 threads (8 waves). Wave w handles frame-slot w,
// lane handles channels [lane*4, lane*4+4). Async double-buffered staging.
// ---------------------------------------------------------------------------
__global__ __launch_bounds__(256) void fn_partial_kernel(
    const float* __restrict__ feat, const int* __restrict__ flen,
    float* __restrict__ part) {
  // [0,2048): two 8x128 staging buffers; [2048,4096): cross-wave reduction
  __shared__ __align__(16) float smem[4096];

  const int b = blockIdx.x / SPLIT;
  const int s = blockIdx.x - b * SPLIT;
  const int w = threadIdx.x >> 5;
  const int lane = threadIdx.x & 31;

  const int len = flen[b];
  int valid = len - s * FPB;
  valid = valid < 0 ? 0 : (valid > FPB ? FPB : valid);
  const int nIter = (valid + FPI - 1) / FPI;

  const unsigned long long sbase = (unsigned long long)(uintptr_t)(
      feat + ((size_t)b * Ln + (size_t)s * FPB) * Cn);

  // Low 32 bits of a flat shared pointer == LDS byte offset (ISA ch.10.2).
  const unsigned lds0 = (unsigned)(uintptr_t)(&smem[0]);
  const unsigned slotB = (unsigned)((w * 32 + lane) * 16); // bytes in a buffer
  const unsigned ldsA = lds0 + slotB;
  const unsigned ldsB = lds0 + 4096u + slotB;

  v4f sum = {0.f, 0.f, 0.f, 0.f};
  v4f sq = {0.f, 0.f, 0.f, 0.f};

  if (nIter > 0) {
    async_ld_b128(ldsA, (unsigned)((w * Cn + lane * 4) * 4), sbase);
    for (int i = 0; i < nIter; ++i) {
      const int buf = i & 1;
      if (i + 1 < nIter) {
        const unsigned voff =
            (unsigned)((((i + 1) * FPI + w) * Cn + lane * 4) * 4);
        async_ld_b128(buf ? ldsA : ldsB, voff, sbase);
        WAIT_ASYNC(1); // previous buffer's copy complete, next in flight
      } else {
        WAIT_ASYNC(0);
      }
      const int frame = i * FPI + w;
      const float m = (frame < valid) ? 1.0f : 0.0f;
      const v4f v =
          *(const v4f*)(&smem[(size_t)buf * 1024 + (w * 32 + lane) * 4]);
      sum += v * m;
      sq += v * v * m;
    }
  }

  // Cross-wave reduce (8 frame-slots) in LDS region [2048,4096).
  *(v4f*)(&smem[2048 + (w * 32 + lane) * 4]) = sum;
  *(v4f*)(&smem[3072 + (w * 32 + lane) * 4]) = sq;
  __syncthreads();
  if (threadIdx.x < 32) {
    v4f S = {0.f, 0.f, 0.f, 0.f};
    v4f Q = {0.f, 0.f, 0.f, 0.f};
    for (int ww = 0; ww < 8; ++ww) {
      S += *(const v4f*)(&smem[2048 + (ww * 32 + threadIdx.x) * 4]);
      Q += *(const v4f*)(&smem[3072 + (ww * 32 + threadIdx.x) * 4]);
    }
    float* dst = part + (size_t)blockIdx.x * 256 + threadIdx.x * 4;
    *(v4f*)dst = S;
    *(v4f*)(dst + 128) = Q;
  }
}

// ---------------------------------------------------------------------------
// Pass 2: fold SPLIT partials -> mean, rstd per (b,c); copy feat_len to tail.
// ---------------------------------------------------------------------------
__global__ __launch_bounds__(128) void fn_finalize_kernel(
    const float* __restrict__ part, const int* __restrict__ flen,
    float* __restrict__ meanA, float* __restrict__ rstdA,
    int* __restrict__ lenOut) {
  const int b = blockIdx.x;
  const int c = threadIdx.x;
  float sum = 0.f, sq = 0.f;
  for (int s = 0; s < SPLIT; ++s) {
    const size_t o = (size_t)(b * SPLIT + s) * 256;
    sum += part[o + c];
    sq += part[o + 128 + c];
  }
  const float n = (float)flen[b];
  const float mean = sum / n;
  float var = (sq - sum * sum / n) / (n - 1.0f);
  var = fmaxf(var, 0.0f);
  const float sd = fmaxf(sqrtf(var), 1e-10f);
  meanA[b * Cn + c] = mean;
  rstdA[b * Cn + c] = 1.0f / sd;
  if (c == 0) lenOut[b] = flen[b]; // second tuple output: feat_len passthrough
}

// ---------------------------------------------------------------------------
// Pass 3: out = (feat - mean) * rstd over all frames. NT stores (write-once).
// ---------------------------------------------------------------------------
__global__ __launch_bounds__(256) void fn_norm_kernel(
    const float* __restrict__ feat, const float* __restrict__ meanA,
    const float* __restrict__ rstdA, float* __restrict__ out) {
  const int b = blockIdx.x >> 9; // 512 chunks of 8 frames per batch row
  const int chunk = blockIdx.x & 511;
  const int w = threadIdx.x >> 5;
  const int lane = threadIdx.x & 31;
  const size_t idx =
      ((size_t)b * Ln + (size_t)chunk * 8 + w) * Cn + (size_t)lane * 4;
  const v4f x = *(const v4f*)(feat + idx);
  const v4f mu = *(const v4f*)(meanA + b * Cn + lane * 4);
  const v4f rs = *(const v4f*)(rstdA + b * Cn + lane * 4);
  const v4f o = (x - mu) * rs;
  __builtin_nontemporal_store(o, (v4f*)(out + idx));
}

extern "C" void kernel_launch(void* const* d_in, const int* in_sizes, int n_in,
                              void* d_out, int out_size, void* d_ws,
                              size_t ws_size, hipStream_t stream) {
  (void)in_sizes; (void)n_in; (void)out_size; (void)ws_size;
  const float* feat = (const float*)d_in[0];
  const int* flen = (const int*)d_in[1];
  float* out = (float*)d_out;

  float* wsf = (float*)d_ws;
  float* part = wsf;                               // Bn*SPLIT*256 = 262144 f
  float* meanA = wsf + (size_t)Bn * SPLIT * 256;   // 8192 f
  float* rstdA = meanA + (size_t)Bn * Cn;          // 8192 f
  int* lenOut = (int*)out + (size_t)Bn * Ln * Cn;  // tuple tail: feat_len

  fn_partial_kernel<<<Bn * SPLIT, 256, 0, stream>>>(feat, flen, part);
  fn_finalize_kernel<<<Bn, 128, 0, stream>>>(part, flen, meanA, rstdA, lenOut);
  fn_norm_kernel<<<Bn * (Ln / 8), 256, 0, stream>>>(feat, meanA, rstdA, out);
}